// HeteroDotProductPredictor_45440753992134
// MI455X (gfx1250) — compile-verified
//
#include <hip/hip_runtime.h>
#include <hip/hip_bf16.h>

// CDNA5 / gfx1250: edge dot-product via fp32 WMMA.
// score[e] = dot(x[src[e]], x[dst[e]]), E = 640000, D = 128, all fp32.
//
// Strategy: 16 edges per wave32. A = 16x128 rows of x[src], B = 128x16
// columns of x[dst]; V_WMMA_F32_16X16X4_F32 accumulates D = A*B over 32
// K-steps; the 16 edge scores are the diagonal of D. Full fp32 precision.
// Node table (51.2 MB) fits in the 192 MB L2, so this is an L2-BW-bound
// gather; the 16x compute overcount is far below the WMMA ceiling.

typedef __attribute__((ext_vector_type(2))) float v2f;
typedef __attribute__((ext_vector_type(8))) float v8f;

#define D_FEAT 128
#define EDGES_PER_TILE 16
#define WAVES_PER_BLOCK 8
#define THREADS_PER_BLOCK (WAVES_PER_BLOCK * 32)

__global__ __launch_bounds__(THREADS_PER_BLOCK)
void edge_dot_wmma_f32(const float* __restrict__ x,
                       const int* __restrict__ src_idx,
                       const int* __restrict__ dst_idx,
                       float* __restrict__ out,
                       int n_edges) {
    const int lane = threadIdx.x & 31;
    const int wave = threadIdx.x >> 5;
    const int tile = blockIdx.x * WAVES_PER_BLOCK + wave;
    const int e0   = tile * EDGES_PER_TILE;
    if (e0 >= n_edges) return;  // wave-uniform; EXEC stays all-1s inside

    // A-matrix (16x4 fp32): lane l -> row m = l&15, K = 2*(l>=16) + {0,1}
    // B-matrix (4x16 fp32): lane l -> col n = l&15, rows k0+2*(l>=16)+{0,1}
    // Both reduce to one aligned float2 load per K-step from the node row.
    const int m    = lane & 15;
    const int half = lane >> 4;  // 0: lanes 0-15, 1: lanes 16-31

    const int s = src_idx[e0 + m];
    const int d = dst_idx[e0 + m];

    const float* __restrict__ aptr = x + (size_t)s * D_FEAT + 2 * half;
    const float* __restrict__ bptr = x + (size_t)d * D_FEAT + 2 * half;

    v8f c = {};
#pragma unroll
    for (int k0 = 0; k0 < D_FEAT; k0 += 4) {
        v2f a = *(const v2f*)(aptr + k0);
        v2f b = *(const v2f*)(bptr + k0);
        // 8 args: (neg_a, A, neg_b, B, c_mod, C, reuse_a, reuse_b)
        c = __builtin_amdgcn_wmma_f32_16x16x4_f32(
            /*neg_a=*/false, a, /*neg_b=*/false, b,
            /*c_mod=*/(short)0, c, /*reuse_a=*/false, /*reuse_b=*/false);
    }

    // Diagonal extraction (C/D layout: VGPR r: lanes 0-15 -> M=r, N=lane;
    // lanes 16-31 -> M=8+r, N=lane-16). D[r][r] lives in lane r, VGPR r;
    // D[8+r][8+r] lives in lane 24+r, VGPR r.
    float val = 0.0f;
    int   eidx = -1;
#pragma unroll
    for (int r = 0; r < 8; ++r) {
        if (lane == r)      { val = c[r]; eidx = r; }
        if (lane == 24 + r) { val = c[r]; eidx = 8 + r; }
    }
    if (eidx >= 0) out[e0 + eidx] = val;
}

extern "C" void kernel_launch(void* const* d_in, const int* in_sizes, int n_in,
                              void* d_out, int out_size, void* d_ws, size_t ws_size,
                              hipStream_t stream) {
    const float* x       = (const float*)d_in[0];
    const int*   src_idx = (const int*)d_in[1];
    const int*   dst_idx = (const int*)d_in[2];
    float*       out     = (float*)d_out;

    const int n_edges = in_sizes[1];                       // 640000
    const int n_tiles = (n_edges + EDGES_PER_TILE - 1) / EDGES_PER_TILE;
    const int n_blocks = (n_tiles + WAVES_PER_BLOCK - 1) / WAVES_PER_BLOCK;

    edge_dot_wmma_f32<<<n_blocks, THREADS_PER_BLOCK, 0, stream>>>(
        x, src_idx, dst_idx, out, n_edges);
}